// AttentiveGRU2_11287174053942
// MI455X (gfx1250) — compile-verified
//
#include <hip/hip_runtime.h>
#include <math.h>

#define N_NODES 50000
#define N_EDGES 800000
#define DFEAT   128

typedef __attribute__((ext_vector_type(2))) float v2f;
typedef __attribute__((ext_vector_type(8))) float v8f;

// ---- order-preserving float<->int mapping for atomic segment-max ----
__device__ __forceinline__ int   orderF(float f) {
    int i = __float_as_int(f);
    return i >= 0 ? i : (i ^ 0x7FFFFFFF);
}
__device__ __forceinline__ float unorderF(int i) {
    return __int_as_float(i >= 0 ? i : (i ^ 0x7FFFFFFF));
}

// ---- 1) init workspace: emax=-inf(ordered), esum=0, c=0 ----
__global__ __launch_bounds__(256) void init_ws_k(int* emax, float* esum, float* c) {
    int idx = blockIdx.x * blockDim.x + threadIdx.x;
    if (idx < N_NODES * DFEAT) c[idx] = 0.0f;
    if (idx < N_NODES) { emax[idx] = (int)0x80000000; esum[idx] = 0.0f; }
}

// ---- 2) segment max over dst ----
__global__ __launch_bounds__(256) void seg_max_k(const float* __restrict__ logits,
                                                 const int* __restrict__ dst,
                                                 int* __restrict__ emax) {
    int e = blockIdx.x * blockDim.x + threadIdx.x;
    if (e < N_EDGES) atomicMax(&emax[dst[e]], orderF(logits[e]));
}

// ---- 3) ex = exp(logit - max[dst]); esum[dst] += ex ----
__global__ __launch_bounds__(256) void seg_expsum_k(const float* __restrict__ logits,
                                                    const int* __restrict__ dst,
                                                    const int* __restrict__ emax,
                                                    float* __restrict__ ex,
                                                    float* __restrict__ esum) {
    int e = blockIdx.x * blockDim.x + threadIdx.x;
    if (e >= N_EDGES) return;
    int d = dst[e];
    float v = expf(logits[e] - unorderF(emax[d]));
    ex[e] = v;
    atomicAdd(&esum[d], v);
}

// ---- 4/7/8) C[m, j] = sum_k A[m,k] * W[j,k] + bias[j]  (fp32 WMMA) ----
// A: [M x 128] row-major, W: [OUT x 128] row-major, C: [M x OUT].
// One wave per 16x16 output tile; K in steps of 4 via V_WMMA_F32_16X16X4_F32.
// A frag (16x4): lanes 0-15 carry M rows with K={k,k+1}; lanes 16-31 K={k+2,k+3}.
// B frag (4x16): lane (n&15) carries column n0+n with the same K-pair split.
__global__ __launch_bounds__(256) void gemm_wmma_bias_k(const float* __restrict__ A,
                                                        const float* __restrict__ W,
                                                        const float* __restrict__ bias,
                                                        float* __restrict__ C,
                                                        int OUT) {
    const int wave = threadIdx.x >> 5;
    const int lane = threadIdx.x & 31;
    const int half = lane >> 4;        // 0: K-pair {0,1}, 1: K-pair {2,3}
    const int l15  = lane & 15;
    const int m0   = blockIdx.x * 16;  // M is a multiple of 16 (50000 = 3125*16)
    const int ntiles = OUT >> 4;

    const float* Ap = A + (m0 + l15) * DFEAT + half * 2;

    for (int jt = wave; jt < ntiles; jt += 8) {
        const int n0 = jt * 16;
        const float* Wp = W + (n0 + l15) * DFEAT + half * 2;
        v8f acc = {};
#pragma unroll
        for (int k = 0; k < DFEAT; k += 4) {
            v2f a = *(const v2f*)(Ap + k);
            v2f b = *(const v2f*)(Wp + k);
            // emits v_wmma_f32_16x16x4_f32
            acc = __builtin_amdgcn_wmma_f32_16x16x4_f32(
                /*neg_a=*/false, a, /*neg_b=*/false, b,
                /*c_mod=*/(short)0, acc, /*reuse_a=*/false, /*reuse_b=*/false);
        }
        const float bj = bias[n0 + l15];
#pragma unroll
        for (int r = 0; r < 8; ++r) {
            // C/D layout: VGPR r -> row m0 + r (lanes 0-15) / m0 + 8 + r (lanes 16-31)
            int row = m0 + r + half * 8;
            C[row * OUT + n0 + l15] = acc[r] + bj;
        }
    }
}

// ---- 5) per-edge: c[dst] += hv[src] * (ex/esum[dst]); one wave per edge ----
__global__ __launch_bounds__(256) void scatter_msg_k(const int* __restrict__ src,
                                                     const int* __restrict__ dst,
                                                     const float* __restrict__ ex,
                                                     const float* __restrict__ esum,
                                                     const float* __restrict__ hv,
                                                     float* __restrict__ c) {
    int gid  = blockIdx.x * blockDim.x + threadIdx.x;
    int e    = gid >> 5;
    int lane = gid & 31;
    if (e >= N_EDGES) return;
    int s = src[e], d = dst[e];
    float a = ex[e] / esum[d];
    float4 v = *((const float4*)(hv + s * DFEAT) + lane);  // 512B coalesced row
    float* cp = c + d * DFEAT + lane * 4;
    atomicAdd(cp + 0, v.x * a);
    atomicAdd(cp + 1, v.y * a);
    atomicAdd(cp + 2, v.z * a);
    atomicAdd(cp + 3, v.w * a);
}

// ---- 6) context = elu(c), in place ----
__global__ __launch_bounds__(256) void elu_k(float* __restrict__ c) {
    int idx = blockIdx.x * blockDim.x + threadIdx.x;
    if (idx >= N_NODES * DFEAT) return;
    float x = c[idx];
    c[idx] = x > 0.0f ? x : (expf(x) - 1.0f);
}

// ---- 9) GRU gates + ReLU ----
__global__ __launch_bounds__(256) void gru_out_k(const float* __restrict__ gi,
                                                 const float* __restrict__ gh,
                                                 const float* __restrict__ hprev,
                                                 float* __restrict__ out) {
    int idx = blockIdx.x * blockDim.x + threadIdx.x;
    if (idx >= N_NODES * DFEAT) return;
    int n = idx >> 7;       // / 128
    int d = idx & 127;
    const float* gin = gi + n * (3 * DFEAT);
    const float* ghn = gh + n * (3 * DFEAT);
    float r  = 1.0f / (1.0f + expf(-(gin[d]       + ghn[d])));
    float z  = 1.0f / (1.0f + expf(-(gin[d + 128] + ghn[d + 128])));
    float nn = tanhf(gin[d + 256] + r * ghn[d + 256]);
    float h  = (1.0f - z) * nn + z * hprev[idx];
    out[idx] = fmaxf(h, 0.0f);
}

extern "C" void kernel_launch(void* const* d_in, const int* in_sizes, int n_in,
                              void* d_out, int out_size, void* d_ws, size_t ws_size,
                              hipStream_t stream) {
    const float* edge_logits = (const float*)d_in[0];
    const float* node_feats  = (const float*)d_in[1];
    const int*   src         = (const int*)d_in[2];
    const int*   dst         = (const int*)d_in[3];
    const float* W_proj      = (const float*)d_in[4];
    const float* b_proj      = (const float*)d_in[5];
    const float* W_ih        = (const float*)d_in[6];
    const float* b_ih        = (const float*)d_in[7];
    const float* W_hh        = (const float*)d_in[8];
    const float* b_hh        = (const float*)d_in[9];
    float* out = (float*)d_out;

    // workspace carve-up
    char* p = (char*)d_ws;
    int*   emax = (int*)p;   p += (size_t)N_NODES * 4;
    float* esum = (float*)p; p += (size_t)N_NODES * 4;
    float* ex   = (float*)p; p += (size_t)N_EDGES * 4;
    float* hv   = (float*)p; p += (size_t)N_NODES * DFEAT * 4;
    float* c    = (float*)p; p += (size_t)N_NODES * DFEAT * 4;
    float* gi   = (float*)p; p += (size_t)N_NODES * 3 * DFEAT * 4;
    float* gh   = (float*)p; p += (size_t)N_NODES * 3 * DFEAT * 4;

    const int ND = N_NODES * DFEAT;

    init_ws_k   <<<(ND + 255) / 256, 256, 0, stream>>>(emax, esum, c);
    seg_max_k   <<<(N_EDGES + 255) / 256, 256, 0, stream>>>(edge_logits, dst, emax);
    seg_expsum_k<<<(N_EDGES + 255) / 256, 256, 0, stream>>>(edge_logits, dst, emax, ex, esum);

    // hv = node_feats @ W_proj^T + b_proj        (OUT = 128)
    gemm_wmma_bias_k<<<N_NODES / 16, 256, 0, stream>>>(node_feats, W_proj, b_proj, hv, 128);

    // c[dst] += hv[src] * a   (wave-per-edge)
    scatter_msg_k<<<(N_EDGES * 32 + 255) / 256, 256, 0, stream>>>(src, dst, ex, esum, hv, c);

    // context = elu(c)
    elu_k<<<(ND + 255) / 256, 256, 0, stream>>>(c);

    // gi = context @ W_ih^T + b_ih ; gh = node_feats @ W_hh^T + b_hh   (OUT = 384)
    gemm_wmma_bias_k<<<N_NODES / 16, 256, 0, stream>>>(c,          W_ih, b_ih, gi, 384);
    gemm_wmma_bias_k<<<N_NODES / 16, 256, 0, stream>>>(node_feats, W_hh, b_hh, gh, 384);

    // gates + ReLU -> d_out
    gru_out_k<<<(ND + 255) / 256, 256, 0, stream>>>(gi, gh, node_feats, out);
}